// LikFunction_14628658610180
// MI455X (gfx1250) — compile-verified
//
#include <hip/hip_runtime.h>
#include <math.h>

typedef __attribute__((ext_vector_type(2))) float v2f;
typedef __attribute__((ext_vector_type(4))) float v4f;
typedef __attribute__((ext_vector_type(8))) float v8f;

#define STAGE1_BLOCKS 4096
#define TPB 256

// Stage 1: streaming sum of (o-x)^2.
// WMMA f32 16x16x4 with B == all-ones is used as a 64-wide fp32 summation
// engine: D[i][j] += sum_k A[i][k]. Summing all 256 accumulator slots gives
// 16x the true total (16 identical columns), corrected exactly by *1/16.
__global__ __launch_bounds__(TPB) void sumsq_stage1(
    const float* __restrict__ o, const float* __restrict__ x,
    float* __restrict__ partial, long long nQuads)
{
    const long long tid    = (long long)blockIdx.x * TPB + threadIdx.x;
    const long long stride = (long long)gridDim.x * TPB;
    const v4f* __restrict__ o4 = (const v4f*)o;
    const v4f* __restrict__ x4 = (const v4f*)x;

    v8f acc = {};                      // WMMA C/D accumulator (8 VGPRs)
    const v2f ones = {1.0f, 1.0f};     // B matrix: 4x16 of 1.0

    // Main loop: trip count is identical for every thread in the grid
    // (mainQuads is a multiple of step), so EXEC is all-ones at the WMMAs.
    const long long step      = 2 * stride;            // 2-way unroll
    const long long mainQuads = (nQuads / step) * step;

    for (long long q = tid; q < mainQuads; q += step) {
        v4f a0 = o4[q];
        v4f b0 = x4[q];
        v4f a1 = o4[q + stride];
        v4f b1 = x4[q + stride];
        v4f d0 = a0 - b0;
        v4f d1 = a1 - b1;
        v4f s0 = d0 * d0;
        v4f s1 = d1 * d1;
        v2f p;
        p.x = s0.x; p.y = s0.y;
        acc = __builtin_amdgcn_wmma_f32_16x16x4_f32(false, p, false, ones,
                                                    (short)0, acc, false, false);
        p.x = s0.z; p.y = s0.w;
        acc = __builtin_amdgcn_wmma_f32_16x16x4_f32(false, p, false, ones,
                                                    (short)0, acc, false, false);
        p.x = s1.x; p.y = s1.y;
        acc = __builtin_amdgcn_wmma_f32_16x16x4_f32(false, p, false, ones,
                                                    (short)0, acc, false, false);
        p.x = s1.z; p.y = s1.w;
        acc = __builtin_amdgcn_wmma_f32_16x16x4_f32(false, p, false, ones,
                                                    (short)0, acc, false, false);
    }

    // Tail (possibly divergent): plain FMA accumulation, outside the WMMA loop.
    float tacc = 0.0f;
    for (long long q = mainQuads + tid; q < nQuads; q += stride) {
        v4f a = o4[q];
        v4f b = x4[q];
        v4f d = a - b;
        tacc = fmaf(d.x, d.x, tacc);
        tacc = fmaf(d.y, d.y, tacc);
        tacc = fmaf(d.z, d.z, tacc);
        tacc = fmaf(d.w, d.w, tacc);
    }

    // Per-lane: collapse the 8 accumulator slots; wave-sum of all 256 slots
    // is 16x the true total, so scale by 1/16 (exact, power of two).
    float t = (acc[0] + acc[1] + acc[2] + acc[3] +
               acc[4] + acc[5] + acc[6] + acc[7]) * 0.0625f + tacc;

    #pragma unroll
    for (int off = 16; off >= 1; off >>= 1)
        t += __shfl_xor(t, off, 32);

    __shared__ float lds[TPB / 32];
    const int lane = threadIdx.x & 31;
    const int wid  = threadIdx.x >> 5;
    if (lane == 0) lds[wid] = t;
    __syncthreads();
    if (threadIdx.x == 0) {
        float s = 0.0f;
        #pragma unroll
        for (int i = 0; i < TPB / 32; ++i) s += lds[i];
        partial[blockIdx.x] = s;
    }
}

// Stage 2: deterministic fixed-order reduction of block partials + closed-form
// log-likelihood. Single block -> bit-reproducible across graph replays.
__global__ __launch_bounds__(TPB) void lik_finalize(
    const float* __restrict__ partial, int nPartials,
    const float* __restrict__ o, const float* __restrict__ x,
    long long remStart, long long nElems,
    const float* __restrict__ noisePtr,
    float* __restrict__ out, long long nRows)
{
    float s = 0.0f;
    for (int i = threadIdx.x; i < nPartials; i += TPB)
        s += partial[i];

    #pragma unroll
    for (int off = 16; off >= 1; off >>= 1)
        s += __shfl_xor(s, off, 32);

    __shared__ float lds[TPB / 32];
    const int lane = threadIdx.x & 31;
    const int wid  = threadIdx.x >> 5;
    if (lane == 0) lds[wid] = s;
    __syncthreads();

    if (threadIdx.x == 0) {
        float sq = 0.0f;
        #pragma unroll
        for (int i = 0; i < TPB / 32; ++i) sq += lds[i];
        // scalar remainder if nElems wasn't a multiple of 4 (not hit for the
        // reference shapes, kept for robustness)
        for (long long e = remStart; e < nElems; ++e) {
            float d = o[e] - x[e];
            sq = fmaf(d, d, sq);
        }
        const float  noise  = noisePtr[0];
        const double log2pi = 1.8378770664093453;  // log(2*pi)
        const double nd     = (double)nElems;      // n * d
        double v = -0.5 * nd * log2pi
                 - 0.5 * (double)nRows * (double)noise
                 - 0.5 * exp(-2.0 * (double)noise) * (double)sq;
        out[0] = (float)v;
    }
}

extern "C" void kernel_launch(void* const* d_in, const int* in_sizes, int n_in,
                              void* d_out, int out_size, void* d_ws, size_t ws_size,
                              hipStream_t stream)
{
    const float* o     = (const float*)d_in[0];
    const float* x     = (const float*)d_in[1];
    const float* noise = (const float*)d_in[2];
    float*       out   = (float*)d_out;

    const long long nElems = (long long)in_sizes[0];   // 65536 * 1024
    const long long nQuads = nElems >> 2;
    const long long nRows  = nElems / 1024;            // d = 1024 per reference

    int blocks = STAGE1_BLOCKS;
    if (ws_size < (size_t)blocks * sizeof(float)) {
        blocks = (int)(ws_size / sizeof(float));
        if (blocks < 1) blocks = 1;
    }
    float* partials = (float*)d_ws;

    sumsq_stage1<<<blocks, TPB, 0, stream>>>(o, x, partials, nQuads);
    lik_finalize<<<1, TPB, 0, stream>>>(partials, blocks, o, x,
                                        nQuads << 2, nElems, noise, out, nRows);
}